// SC_att_gen_12893491822790
// MI455X (gfx1250) — compile-verified
//
#include <hip/hip_runtime.h>
#include <hip/hip_bf16.h>

typedef float v2f __attribute__((ext_vector_type(2)));
typedef float v8f __attribute__((ext_vector_type(8)));

#define NB   16
#define CC   512
#define SS   4
#define HW   1024
#define COLS 64      // NB*SS
#define CS   2048    // CC*SS

// ---------------------------------------------------------------------------
// Kernel 1: mean+max pool over H*W=1024 for each (n,c,s); write P[c][n*4+s]
// Pure bandwidth: 134 MB read.
// ---------------------------------------------------------------------------
__global__ void SC_pool_kernel(const float* __restrict__ x, float* __restrict__ P) {
    const int g = blockIdx.x;            // ((n*512 + c)*4 + s)
    const int t = threadIdx.x;           // 256 threads
    const float4* xe = (const float4*)(x + (size_t)g * HW);
    float4 v = xe[t];
    float s = v.x + v.y + v.z + v.w;
    float m = fmaxf(fmaxf(v.x, v.y), fmaxf(v.z, v.w));

    __shared__ float rs[256];
    __shared__ float rm[256];
    rs[t] = s; rm[t] = m;
    __syncthreads();
    for (int off = 128; off > 0; off >>= 1) {
        if (t < off) {
            rs[t] += rs[t + off];
            rm[t] = fmaxf(rm[t], rm[t + off]);
        }
        __syncthreads();
    }
    if (t == 0) {
        int n = g >> 11;
        int c = (g >> 2) & (CC - 1);
        int si = g & 3;
        P[c * COLS + n * SS + si] = rs[0] * (1.0f / (float)HW) + rm[0];
    }
}

// ---------------------------------------------------------------------------
// Kernel 2: Q = wq * P, K = wk * P  via V_WMMA_F32_16X16X4_F32 (f32 WMMA).
// M=512, N=64, K=512. One wave per 16x16 output tile; two accumulators
// (Q and K) sharing the B-tile loads. Results stored in [n][o*4+s] layout.
// ---------------------------------------------------------------------------
__global__ void SC_gemm_wmma_kernel(const float* __restrict__ wq,
                                    const float* __restrict__ wk,
                                    const float* __restrict__ P,
                                    float* __restrict__ Qn,
                                    float* __restrict__ Kn) {
    const int tile = blockIdx.x;         // 128 tiles: 32 M-tiles x 4 N-tiles
    const int tn = tile & 3;
    const int tm = tile >> 2;
    const int lane = threadIdx.x;        // 32 (one full wave; EXEC all ones)
    const int lh   = lane >> 4;          // 0: lanes 0-15, 1: lanes 16-31
    const int l15  = lane & 15;
    const int m0 = tm * 16;
    const int n0 = tn * 16;

    v8f accq = {};
    v8f acck = {};

    // A: lane holds M = m0 + l15; VGPR v holds K = k0 + v + 2*lh (contig pair)
    const float* aq = wq + (size_t)(m0 + l15) * CC + 2 * lh;
    const float* ak = wk + (size_t)(m0 + l15) * CC + 2 * lh;
    // B: lane holds N = n0 + l15; VGPR v holds K = k0 + v + 2*lh
    const float* bb = P + (size_t)(2 * lh) * COLS + n0 + l15;

    for (int k0 = 0; k0 < CC; k0 += 4) {
        v2f a_q = *(const v2f*)(aq + k0);
        v2f a_k = *(const v2f*)(ak + k0);
        v2f b;
        b.x = bb[(size_t)k0 * COLS];
        b.y = bb[(size_t)(k0 + 1) * COLS];
        accq = __builtin_amdgcn_wmma_f32_16x16x4_f32(
            false, a_q, false, b, (short)0, accq, false, false);
        acck = __builtin_amdgcn_wmma_f32_16x16x4_f32(
            false, a_k, false, b, (short)0, acck, false, false);
    }

    // C/D layout: VGPR r, lanes 0-15 -> M = r, lanes 16-31 -> M = r + 8
    const int Ng = n0 + l15;         // column = n*4 + s
    const int nn = Ng >> 2;
    const int ss = Ng & 3;
    float* qrow = Qn + (size_t)nn * CS + ss;
    float* krow = Kn + (size_t)nn * CS + ss;
#pragma unroll
    for (int r = 0; r < 8; ++r) {
        int Mg = m0 + r + (lh << 3);
        qrow[(size_t)Mg * SS] = accq[r];
        krow[(size_t)Mg * SS] = acck[r];
    }
}

// ---------------------------------------------------------------------------
// Kernel 3: attention[n,i,j] = softmax_j( q[n,i] * k[n,j] ).
// Row max = qi>=0 ? qi*kmax : qi*kmin (rank-1 trick -> one reduce per n).
// Each block: cache k[n,:] (8KB LDS), then 16 rows; 268 MB coalesced writes.
// ---------------------------------------------------------------------------
__global__ void SC_softmax_kernel(const float* __restrict__ Qn,
                                  const float* __restrict__ Kn,
                                  float* __restrict__ out) {
    const int b = blockIdx.x;            // 16 n * 128 row-groups
    const int n = b >> 7;
    const int rowBase = (b & 127) * 16;
    const int t = threadIdx.x;           // 256 threads

    __shared__ float kk[CS];
    __shared__ float red[256];

    const float4* kv = (const float4*)(Kn + (size_t)n * CS);
    float4 k0 = kv[t * 2];
    float4 k1 = kv[t * 2 + 1];
    ((float4*)kk)[t * 2]     = k0;
    ((float4*)kk)[t * 2 + 1] = k1;

    float lmax = fmaxf(fmaxf(fmaxf(k0.x, k0.y), fmaxf(k0.z, k0.w)),
                       fmaxf(fmaxf(k1.x, k1.y), fmaxf(k1.z, k1.w)));
    float lmin = fminf(fminf(fminf(k0.x, k0.y), fminf(k0.z, k0.w)),
                       fminf(fminf(k1.x, k1.y), fminf(k1.z, k1.w)));

    red[t] = lmax; __syncthreads();
    for (int off = 128; off > 0; off >>= 1) {
        if (t < off) red[t] = fmaxf(red[t], red[t + off]);
        __syncthreads();
    }
    const float kmax = red[0]; __syncthreads();

    red[t] = lmin; __syncthreads();
    for (int off = 128; off > 0; off >>= 1) {
        if (t < off) red[t] = fminf(red[t], red[t + off]);
        __syncthreads();
    }
    const float kmin = red[0]; __syncthreads();

    const float* q = Qn + (size_t)n * CS;
    const size_t outBase = ((size_t)n * CS + (size_t)rowBase) * CS;

    for (int r = 0; r < 16; ++r) {
        const float qi = q[rowBase + r];
        const float m = (qi >= 0.0f) ? qi * kmax : qi * kmin;
        float e[8];
        float ls = 0.0f;
#pragma unroll
        for (int u = 0; u < 8; ++u) {
            e[u] = __expf(qi * kk[t * 8 + u] - m);
            ls += e[u];
        }
        red[t] = ls; __syncthreads();
        for (int off = 128; off > 0; off >>= 1) {
            if (t < off) red[t] += red[t + off];
            __syncthreads();
        }
        const float inv = 1.0f / red[0];
        __syncthreads();                 // red reused next iteration

        float4 o0 = { e[0] * inv, e[1] * inv, e[2] * inv, e[3] * inv };
        float4 o1 = { e[4] * inv, e[5] * inv, e[6] * inv, e[7] * inv };
        float4* orow = (float4*)(out + outBase + (size_t)r * CS);
        orow[t * 2]     = o0;
        orow[t * 2 + 1] = o1;
    }
}

extern "C" void kernel_launch(void* const* d_in, const int* in_sizes, int n_in,
                              void* d_out, int out_size, void* d_ws, size_t ws_size,
                              hipStream_t stream) {
    (void)in_sizes; (void)n_in; (void)out_size; (void)ws_size;
    const float* x  = (const float*)d_in[0];
    const float* wq = (const float*)d_in[1];
    const float* wk = (const float*)d_in[2];
    float* out = (float*)d_out;

    float* P  = (float*)d_ws;            // [512][64]   = 128 KB
    float* Qn = P  + CC * COLS;          // [16][2048]  = 128 KB
    float* Kn = Qn + NB * CS;            // [16][2048]  = 128 KB

    SC_pool_kernel<<<NB * CC * SS, 256, 0, stream>>>(x, P);
    SC_gemm_wmma_kernel<<<(CC / 16) * (COLS / 16), 32, 0, stream>>>(wq, wk, P, Qn, Kn);
    SC_softmax_kernel<<<NB * (CS / 16), 256, 0, stream>>>(Qn, Kn, out);
}